// GraphConvLayer_2637109919861
// MI455X (gfx1250) — compile-verified
//
#include <hip/hip_runtime.h>
#include <hip/hip_bf16.h>

// GraphConv: h = relu( scatter_sum(feature[src] -> dst) @ W.T + b )
// N=50000 nodes, E=800000 edges, D_IN=D_OUT=128, all fp32.

typedef __attribute__((ext_vector_type(2))) float v2f;
typedef __attribute__((ext_vector_type(8))) float v8f;

#define D 128
#define LDS_STRIDE 132   // 128 + 4 pad: kills LDS bank conflicts on A-fragment reads

// ---------------------------------------------------------------------------
// Kernel 1: zero the aggregation workspace (float4-vectorized)
// ---------------------------------------------------------------------------
__global__ void gc_zero_kernel(float4* __restrict__ p, int n4) {
    int i = blockIdx.x * blockDim.x + threadIdx.x;
    if (i < n4) p[i] = make_float4(0.f, 0.f, 0.f, 0.f);
}

// ---------------------------------------------------------------------------
// Kernel 2: scatter-sum.  One wave32 per edge; lane L owns feature bytes
// [L*16, L*16+16) of the 512-byte row -> float4 load + 4 atomic f32 adds.
// src[e]/dst[e] are wave-uniform -> scalar loads.
// ---------------------------------------------------------------------------
__global__ void gc_scatter_kernel(const float* __restrict__ feature,
                                  const int* __restrict__ src,
                                  const int* __restrict__ dst,
                                  float* __restrict__ agg,
                                  int nEdges) {
    long idx = (long)blockIdx.x * blockDim.x + threadIdx.x;
    int e    = (int)(idx >> 5);
    int lane = (int)(idx & 31);
    if (e >= nEdges) return;

    int s = src[e];
    int d = dst[e];

    float4 v = *(const float4*)(feature + (size_t)s * D + lane * 4);
    float* dp = agg + (size_t)d * D + lane * 4;
    __hip_atomic_fetch_add(dp + 0, v.x, __ATOMIC_RELAXED, __HIP_MEMORY_SCOPE_AGENT);
    __hip_atomic_fetch_add(dp + 1, v.y, __ATOMIC_RELAXED, __HIP_MEMORY_SCOPE_AGENT);
    __hip_atomic_fetch_add(dp + 2, v.z, __ATOMIC_RELAXED, __HIP_MEMORY_SCOPE_AGENT);
    __hip_atomic_fetch_add(dp + 3, v.w, __ATOMIC_RELAXED, __HIP_MEMORY_SCOPE_AGENT);
}

// ---------------------------------------------------------------------------
// Kernel 3: out = relu(agg @ W.T + b) with V_WMMA_F32_16X16X4_F32.
// Block = 256 threads = 8 waves; block owns 16 rows, wave w owns cols
// [16w, 16w+16).  A tile (16x128) staged in LDS once; K-loop = 32 WMMAs.
//
// f32 16x16x4 fragment layouts (ISA 7.12.2):
//   A (16x4):  lane L, vgpr v -> M = L%16, K = (L/16)*2 + v
//   B (4x16):  lane L, vgpr v -> N = L%16, K = (L/16)*2 + v   (B[k][n] = W[n][k])
//   C/D:       lane L, vgpr r -> M = r + 8*(L/16), N = L%16
// ---------------------------------------------------------------------------
__global__ void gc_gemm_kernel(const float* __restrict__ agg,
                               const float* __restrict__ W,
                               const float* __restrict__ bias,
                               float* __restrict__ out,
                               int nNodes) {
    __shared__ float Atile[16 * LDS_STRIDE];

    const int row0 = blockIdx.x * 16;

    // Cooperative A-tile load: 16 rows x 32 float4 = 512 items over 256 threads.
    for (int i = threadIdx.x; i < 16 * (D / 4); i += blockDim.x) {
        int r = i >> 5;          // row in tile
        int c = i & 31;          // float4 column
        int grow = row0 + r;
        float4 v = make_float4(0.f, 0.f, 0.f, 0.f);
        if (grow < nNodes)
            v = *(const float4*)(agg + (size_t)grow * D + c * 4);
        *(float4*)&Atile[r * LDS_STRIDE + c * 4] = v;
    }
    __syncthreads();

    const int wave = threadIdx.x >> 5;
    const int lane = threadIdx.x & 31;
    const int col0 = wave * 16;          // this wave's 16-col output tile
    const int mn   = lane & 15;          // M for A frags / N for B frags
    const int kh   = (lane >> 4) * 2;    // K sub-block: 0 or 2

    const float* Wn = W + (size_t)(col0 + mn) * D;  // row of W == col of B

    v8f c = {};
    #pragma unroll
    for (int k = 0; k < D; k += 4) {
        v2f a, b;
        a[0] = Atile[mn * LDS_STRIDE + k + kh + 0];
        a[1] = Atile[mn * LDS_STRIDE + k + kh + 1];
        b[0] = Wn[k + kh + 0];
        b[1] = Wn[k + kh + 1];
        // D = A*B + C  (fp32, full precision)
        c = __builtin_amdgcn_wmma_f32_16x16x4_f32(
                /*neg_a=*/false, a, /*neg_b=*/false, b,
                /*c_mod=*/(short)0, c, /*reuse_a=*/false, /*reuse_b=*/false);
    }

    const float bn = bias[col0 + mn];
    const int mbase = (lane >> 4) << 3;  // 0 or 8
    #pragma unroll
    for (int r = 0; r < 8; ++r) {
        int grow = row0 + mbase + r;
        float v = c[r] + bn;
        v = v > 0.f ? v : 0.f;
        if (grow < nNodes)
            out[(size_t)grow * D + col0 + mn] = v;
    }
}

// ---------------------------------------------------------------------------
extern "C" void kernel_launch(void* const* d_in, const int* in_sizes, int n_in,
                              void* d_out, int out_size, void* d_ws, size_t ws_size,
                              hipStream_t stream) {
    const float* feature    = (const float*)d_in[0];  // [N, 128]
    const float* W          = (const float*)d_in[1];  // [128, 128]
    const float* bias       = (const float*)d_in[2];  // [128]
    const int*   edge_index = (const int*)d_in[3];    // [2, E] flat

    const int nNodes = in_sizes[0] / D;
    const int nEdges = in_sizes[3] / 2;
    const int* src = edge_index;
    const int* dst = edge_index + nEdges;

    float* agg = (float*)d_ws;               // [N, 128] fp32 accumulator
    float* out = (float*)d_out;

    // 1) agg = 0
    int n4 = nNodes * (D / 4);
    gc_zero_kernel<<<(n4 + 255) / 256, 256, 0, stream>>>((float4*)agg, n4);

    // 2) scatter-sum: one wave per edge
    long items = (long)nEdges * 32;
    int blocks = (int)((items + 255) / 256);
    gc_scatter_kernel<<<blocks, 256, 0, stream>>>(feature, src, dst, agg, nEdges);

    // 3) WMMA GEMM + bias + ReLU
    int gblocks = (nNodes + 15) / 16;
    gc_gemm_kernel<<<gblocks, 256, 0, stream>>>(agg, W, bias, out, nNodes);
}